// PiNet_87754771792351
// MI455X (gfx1250) — compile-verified
//
#include <hip/hip_runtime.h>

#define NNODES 100000
#define NEDGES 3200000
#define NOUT   10

typedef __attribute__((ext_vector_type(2))) float v2f;
typedef __attribute__((ext_vector_type(8))) float v8f;

// ---------------- degree / dinv ----------------
__global__ void degree_count(const int* __restrict__ dst, int* __restrict__ deg) {
    int e = blockIdx.x * blockDim.x + threadIdx.x;
    if (e < NEDGES) atomicAdd(&deg[dst[e]], 1);
}

__global__ void make_dinv(const int* __restrict__ deg, float* __restrict__ dinv) {
    int i = blockIdx.x * blockDim.x + threadIdx.x;
    if (i < NNODES) dinv[i] = rsqrtf((float)deg[i] + 1.0f);   // +1 = self loop
}

__global__ void init_bias(const float* __restrict__ b, float* __restrict__ A) {
    int i = blockIdx.x * blockDim.x + threadIdx.x;            // N*64 elements
    if (i < NNODES * 64) A[i] = b[i & 63];
}

// ---------------- H = X(N x K) @ W(K x 64), f32 WMMA ----------------
// block = 128 threads (4 waves). block covers 16 rows x 64 cols.
// wave w owns the 16x16 tile at columns w*16.
template <int K>
__global__ void gemm_xw(const float* __restrict__ X, const float* __restrict__ W,
                        float* __restrict__ H) {
    __shared__ float Wl[K * 64];
    __shared__ float Xl[16 * K];
    const int tid = threadIdx.x;
    const int r0  = blockIdx.x * 16;

    for (int i = tid; i < K * 64; i += 128) Wl[i] = W[i];
    for (int i = tid; i < 16 * K; i += 128) {
        int r = i / K, c = i % K;
        Xl[i] = X[(size_t)(r0 + r) * K + c];
    }
    __syncthreads();

    const int wave = tid >> 5;
    const int lane = tid & 31;
    const int n0   = wave * 16;
    const int m    = lane & 15;            // row (A) / col (B,C) within tile
    const int koff = (lane >> 4) * 2;      // lanes 16-31 hold K+2,K+3

    v8f c = {};
#pragma unroll
    for (int k0 = 0; k0 < K; k0 += 4) {
        v2f a, b;
        a.x = Xl[m * K + k0 + koff];
        a.y = Xl[m * K + k0 + koff + 1];
        b.x = Wl[(k0 + koff) * 64 + n0 + m];
        b.y = Wl[(k0 + koff + 1) * 64 + n0 + m];
        c = __builtin_amdgcn_wmma_f32_16x16x4_f32(false, a, false, b,
                                                  (short)0, c, false, false);
    }
#pragma unroll
    for (int i = 0; i < 8; ++i) {
        int row = r0 + (lane >> 4) * 8 + i;
        int col = n0 + m;
        H[(size_t)row * 64 + col] = c[i];
    }
}

// ---------------- scatter: A[dst] += H[src] * norm  (one wave per edge) ----------------
__global__ void scatter_msg(const float* __restrict__ H, const int* __restrict__ src,
                            const int* __restrict__ dst, const float* __restrict__ dinv,
                            float* __restrict__ A) {
    const int gid  = blockIdx.x * blockDim.x + threadIdx.x;
    const int wid  = gid >> 5;
    const int lane = gid & 31;
    const int total = NEDGES + NNODES;
    if (wid >= total) return;
    int s, d;
    if (wid < NEDGES) { s = src[wid]; d = dst[wid]; }
    else              { s = wid - NEDGES; d = s; }           // self loop
    const float nrm = dinv[s] * dinv[d];
    const float2 h2 = *reinterpret_cast<const float2*>(H + (size_t)s * 64 + lane * 2);
    float* out = A + (size_t)d * 64 + lane * 2;
    atomicAdd(out + 0, h2.x * nrm);
    atomicAdd(out + 1, h2.y * nrm);
}

// ---------------- G(64x64) = A2^T @ A2, f32 WMMA, K = N ----------------
// grid.x = 16 tiles (4x4), grid.y = 125 chunks of 800 rows; block = 128 (4 waves).
__global__ void gemm_ata(const float* __restrict__ A2, float* __restrict__ G) {
    const int tile = blockIdx.x;
    const int tm   = (tile >> 2) * 16;
    const int tn   = (tile & 3) * 16;
    const int base = blockIdx.y * 800;
    const int wave = threadIdx.x >> 5;
    const int lane = threadIdx.x & 31;
    const int m    = lane & 15;
    const int koff = (lane >> 4) * 2;

    v8f c = {};
    for (int k0 = base + wave * 4; k0 < base + 800; k0 += 16) {
        const float* r0p = A2 + (size_t)(k0 + koff) * 64;
        const float* r1p = A2 + (size_t)(k0 + koff + 1) * 64;
        v2f a, b;
        a.x = r0p[tm + m];  a.y = r1p[tm + m];   // A = A2^T tile
        b.x = r0p[tn + m];  b.y = r1p[tn + m];   // B = A2 tile
        c = __builtin_amdgcn_wmma_f32_16x16x4_f32(false, a, false, b,
                                                  (short)0, c, false, false);
    }
#pragma unroll
    for (int i = 0; i < 8; ++i) {
        int row = tm + (lane >> 4) * 8 + i;
        int col = tn + m;
        atomicAdd(&G[row * 64 + col], c[i]);
    }
}

// ---------------- o = vec(G) @ Wl + bl; softmax ----------------
__global__ void head(const float* __restrict__ G, const float* __restrict__ Wl,
                     const float* __restrict__ bl, float* __restrict__ out) {
    __shared__ float sh[NOUT];
    const int wave = threadIdx.x >> 5;   // 10 waves, block = 320
    const int lane = threadIdx.x & 31;
    float acc = 0.f;
    for (int i = lane; i < 4096; i += 32)
        acc += G[i] * Wl[i * NOUT + wave];
#pragma unroll
    for (int off = 16; off; off >>= 1)
        acc += __shfl_xor(acc, off, 32);
    if (lane == 0) sh[wave] = acc + bl[wave];
    __syncthreads();
    if (threadIdx.x == 0) {
        float mx = sh[0];
        for (int j = 1; j < NOUT; ++j) mx = fmaxf(mx, sh[j]);
        float s = 0.f, e[NOUT];
        for (int j = 0; j < NOUT; ++j) { e[j] = __expf(sh[j] - mx); s += e[j]; }
        for (int j = 0; j < NOUT; ++j) out[j] = e[j] / s;
    }
}

extern "C" void kernel_launch(void* const* d_in, const int* in_sizes, int n_in,
                              void* d_out, int out_size, void* d_ws, size_t ws_size,
                              hipStream_t stream) {
    const float* x    = (const float*)d_in[0];
    const int*   ei   = (const int*)d_in[1];
    const int*   srcp = ei;            // edge_index[0]
    const int*   dstp = ei + NEDGES;   // edge_index[1]
    const float* Wa1  = (const float*)d_in[2];
    const float* ba1  = (const float*)d_in[3];
    const float* Wa2  = (const float*)d_in[4];
    const float* ba2  = (const float*)d_in[5];
    const float* Wl   = (const float*)d_in[6];
    const float* bl   = (const float*)d_in[7];
    float* out = (float*)d_out;

    char* ws = (char*)d_ws;
    int*   deg  = (int*)(ws);
    float* dinv = (float*)(ws + ((size_t)1 << 19));
    float* H    = (float*)(ws + ((size_t)1 << 20));                       // N*64 f32
    float* A1   = (float*)(ws + ((size_t)1 << 20) + ((size_t)26 << 20));
    float* A2   = (float*)(ws + ((size_t)1 << 20) + ((size_t)52 << 20));
    float* G    = (float*)(ws + ((size_t)1 << 20) + ((size_t)78 << 20));

    hipMemsetAsync(deg, 0, NNODES * sizeof(int), stream);
    hipMemsetAsync(G, 0, 64 * 64 * sizeof(float), stream);

    degree_count<<<(NEDGES + 255) / 256, 256, 0, stream>>>(dstp, deg);
    make_dinv<<<(NNODES + 255) / 256, 256, 0, stream>>>(deg, dinv);

    const int scatter_threads = (NEDGES + NNODES) * 32;
    const int scatter_blocks  = (scatter_threads + 255) / 256;

    // layer 1
    gemm_xw<128><<<NNODES / 16, 128, 0, stream>>>(x, Wa1, H);
    init_bias<<<(NNODES * 64 + 255) / 256, 256, 0, stream>>>(ba1, A1);
    scatter_msg<<<scatter_blocks, 256, 0, stream>>>(H, srcp, dstp, dinv, A1);

    // layer 2
    gemm_xw<64><<<NNODES / 16, 128, 0, stream>>>(A1, Wa2, H);
    init_bias<<<(NNODES * 64 + 255) / 256, 256, 0, stream>>>(ba2, A2);
    scatter_msg<<<scatter_blocks, 256, 0, stream>>>(H, srcp, dstp, dinv, A2);

    // readout
    dim3 gAta(16, 125);
    gemm_ata<<<gAta, 128, 0, stream>>>(A2, G);
    head<<<1, 320, 0, stream>>>(G, Wl, bl, out);
}